// ResBlock_11802570130362
// MI455X (gfx1250) — compile-verified
//
#include <hip/hip_runtime.h>

#define BATCH 8
#define CIN   128
#define CNECK 128
#define COUTC 256
#define NVERT 10242
#define NFACE 20480
#define KDIM  128   // all GEMM K-panels are 128

typedef __attribute__((ext_vector_type(2))) float v2f;
typedef __attribute__((ext_vector_type(8))) float v8f;

// ---------------------------------------------------------------------------
// Generic fp32 WMMA GEMM:  Out[b,m,n] = sum_p sum_k W_p[m,k] * X_p[b,k,n] (+bias[m])
//   W element = W[p*wps + m*wms + k*wks]   (supports transposed/strided coeffs)
//   X panel p = X + p*xps, laid out [B, KDIM, N]
// Block: 256 threads = 8 waves; block tile = 128(M) x 64(N); wave tile = 16x64.
// A strip preloaded to registers per panel; X tile staged in LDS transposed
// [n][k] with stride 132 so each B fragment is one aligned ds_load_b64 and the
// two half-wave K-offsets (+0/+2) hit disjoint bank groups (132 % 64 == 4).
// ---------------------------------------------------------------------------
__global__ __launch_bounds__(256)
void gemm_wmma_f32(const float* __restrict__ W, long wms, long wks, long wps,
                   const float* __restrict__ X, long xps,
                   const float* __restrict__ bias,
                   float* __restrict__ Out,
                   int M, int N, int npanels)
{
    __shared__ float lds[64 * 132];
    const int tid  = threadIdx.x;
    const int lane = tid & 31;
    const int wave = tid >> 5;
    const int b     = blockIdx.z;
    const int nBase = blockIdx.x * 64;
    const int mBase = blockIdx.y * 128 + wave * 16;

    const v8f vzero = {0.f,0.f,0.f,0.f,0.f,0.f,0.f,0.f};
    v8f acc[4];
    #pragma unroll
    for (int nt = 0; nt < 4; ++nt) acc[nt] = vzero;

    const int mrow = mBase + (lane & 15);     // A frag: M = lane&15 (both halves)
    const int koff = (lane >> 4) << 1;        // A/B frag: K offset 0 or 2 per half
    const bool fulltile = (nBase + 64 <= N);

    for (int p = 0; p < npanels; ++p) {
        const float* Xp = X + (size_t)p * xps + (size_t)b * KDIM * (size_t)N;
        const float* Wp = W + (size_t)p * wps;

        // Preload the whole A strip (16 x 128) into registers: 2 floats per
        // lane per k-step -> 32 x v2f. Keeps the hot loop free of VMEM.
        v2f areg[32];
        {
            const float* wr = Wp + (size_t)mrow * wms + (size_t)koff * wks;
            #pragma unroll
            for (int j = 0; j < 32; ++j) {
                areg[j].x = wr[0];
                areg[j].y = wr[wks];
                wr += 4 * wks;
            }
        }

        __syncthreads();
        if (fulltile) {
            // 128 K-rows x 64 N-cols; float4 global loads, transposed LDS store
            for (int idx = tid; idx < 128 * 16; idx += 256) {
                int k  = idx >> 4;
                int n4 = (idx & 15) << 2;
                const float4 v = *(const float4*)(Xp + (size_t)k * N + nBase + n4);
                lds[(n4 + 0) * 132 + k] = v.x;
                lds[(n4 + 1) * 132 + k] = v.y;
                lds[(n4 + 2) * 132 + k] = v.z;
                lds[(n4 + 3) * 132 + k] = v.w;
            }
        } else {
            for (int idx = tid; idx < 128 * 64; idx += 256) {
                int k = idx >> 6;
                int n = idx & 63;
                int gn = nBase + n;
                float v = 0.f;
                if (gn < N) v = Xp[(size_t)k * N + gn];
                lds[n * 132 + k] = v;
            }
        }
        __syncthreads();

        // Prefetch next panel's tile into cache while this panel computes.
        if (p + 1 < npanels) {
            const float* Xn = X + (size_t)(p + 1) * xps + (size_t)b * KDIM * (size_t)N;
            int pr = tid >> 1;                       // 128 rows
            int pc = nBase + ((tid & 1) << 5);       // 2 x 128B segments per row
            if (pc < N) __builtin_prefetch(Xn + (size_t)pr * N + pc, 0, 3);
        }

        const float* lrow = &lds[(lane & 15) * 132 + koff];
        #pragma unroll
        for (int k0 = 0; k0 < KDIM; k0 += 4) {
            const v2f a = areg[k0 >> 2];
            #pragma unroll
            for (int nt = 0; nt < 4; ++nt) {
                v2f bf = *(const v2f*)(lrow + nt * 16 * 132 + k0);
                acc[nt] = __builtin_amdgcn_wmma_f32_16x16x4_f32(
                    false, a, false, bf, (short)0, acc[nt], false, false);
            }
        }
    }

    const int nlocal = lane & 15;
    const int mhalf  = (lane >> 4) << 3;      // C/D: M = g + 8*(lane>=16)
    #pragma unroll
    for (int nt = 0; nt < 4; ++nt) {
        int gn = nBase + nt * 16 + nlocal;
        if (gn < N) {
            #pragma unroll
            for (int g = 0; g < 8; ++g) {
                int m = mBase + mhalf + g;
                float bb = bias ? bias[m] : 0.f;
                Out[((size_t)b * M + m) * N + gn] = acc[nt][g] + bb;
            }
        }
    }
}

// ---------------------------------------------------------------------------
// Per-channel training-mode BN stats over (B, N): emit scale/shift pairs.
// ---------------------------------------------------------------------------
__global__ __launch_bounds__(256)
void bn_stats(const float* __restrict__ y, int C, int N,
              const float* __restrict__ gamma, const float* __restrict__ beta,
              float* __restrict__ stats)
{
    __shared__ float sh[512];
    const int c = blockIdx.x;
    const int tid = threadIdx.x;
    float s = 0.f, s2 = 0.f;
    for (int b = 0; b < BATCH; ++b) {
        const float* p = y + ((size_t)b * C + c) * N;
        for (int n = tid; n < N; n += 256) {
            float v = p[n];
            s += v; s2 += v * v;
        }
    }
    sh[tid] = s; sh[256 + tid] = s2;
    __syncthreads();
    for (int st = 128; st > 0; st >>= 1) {
        if (tid < st) { sh[tid] += sh[tid + st]; sh[256 + tid] += sh[256 + tid + st]; }
        __syncthreads();
    }
    if (tid == 0) {
        float cnt  = (float)(BATCH) * (float)N;
        float mean = sh[0] / cnt;
        float var  = sh[256] / cnt - mean * mean;
        float sc   = gamma[c] * rsqrtf(var + 1e-5f);
        stats[2 * c]     = sc;
        stats[2 * c + 1] = beta[c] - mean * sc;
    }
}

__global__ void bn_apply_relu(float* __restrict__ y, const float* __restrict__ stats,
                              int C, int N, int total)
{
    int i = blockIdx.x * blockDim.x + threadIdx.x;
    if (i >= total) return;
    int c = (i / N) % C;
    float v = fmaf(y[i], stats[2 * c], stats[2 * c + 1]);
    y[i] = v > 0.f ? v : 0.f;
}

// ---------------------------------------------------------------------------
// Face stage: gradient (G: ELL-3 over 3*NF rows) fused with EW/NS dot.
// ---------------------------------------------------------------------------
__global__ void face_grad(const float* __restrict__ h,
                          const int* __restrict__ Gc, const float* __restrict__ Gv,
                          const float* __restrict__ EW, const float* __restrict__ NS,
                          float* __restrict__ gf_ew, float* __restrict__ gf_ns)
{
    int i = blockIdx.x * blockDim.x + threadIdx.x;
    if (i >= BATCH * CNECK * NFACE) return;
    int f  = i % NFACE;
    int bc = i / NFACE;
    const float* hp = h + (size_t)bc * NVERT;
    float ew = 0.f, ns = 0.f;
    #pragma unroll
    for (int d = 0; d < 3; ++d) {
        int base = (d * NFACE + f) * 3;
        float g = 0.f;
        #pragma unroll
        for (int k = 0; k < 3; ++k) g = fmaf(Gv[base + k], hp[Gc[base + k]], g);
        ew = fmaf(g, EW[f * 3 + d], ew);
        ns = fmaf(g, NS[f * 3 + d], ns);
    }
    gf_ew[(size_t)bc * NFACE + f] = ew;
    gf_ns[(size_t)bc * NFACE + f] = ns;
}

// ---------------------------------------------------------------------------
// Vertex stage: laplacian (L: ELL-7) + F2V (ELL-6) on both gradient fields.
// ---------------------------------------------------------------------------
__global__ void vertex_feats(const float* __restrict__ h,
                             const int* __restrict__ Lc, const float* __restrict__ Lv,
                             const int* __restrict__ Fc, const float* __restrict__ Fv,
                             const float* __restrict__ gf_ew, const float* __restrict__ gf_ns,
                             float* __restrict__ lap, float* __restrict__ gv_ew,
                             float* __restrict__ gv_ns)
{
    int i = blockIdx.x * blockDim.x + threadIdx.x;
    if (i >= BATCH * CNECK * NVERT) return;
    int v  = i % NVERT;
    int bc = i / NVERT;
    const float* hp = h + (size_t)bc * NVERT;
    float l = 0.f;
    #pragma unroll
    for (int k = 0; k < 7; ++k) l = fmaf(Lv[v * 7 + k], hp[Lc[v * 7 + k]], l);
    const float* fe = gf_ew + (size_t)bc * NFACE;
    const float* fn = gf_ns + (size_t)bc * NFACE;
    float e = 0.f, n2 = 0.f;
    #pragma unroll
    for (int k = 0; k < 6; ++k) {
        int c = Fc[v * 6 + k];
        float w = Fv[v * 6 + k];
        e  = fmaf(w, fe[c], e);
        n2 = fmaf(w, fn[c], n2);
    }
    lap[i] = l; gv_ew[i] = e; gv_ns[i] = n2;
}

// ---------------------------------------------------------------------------
// Fused residual epilogue: relu(bn3a(x1) + bn1b(y1b))
// ---------------------------------------------------------------------------
__global__ void final_res(const float* __restrict__ x1, const float* __restrict__ y1b,
                          const float* __restrict__ st3, const float* __restrict__ st1b,
                          float* __restrict__ out)
{
    int i = blockIdx.x * blockDim.x + threadIdx.x;
    if (i >= BATCH * COUTC * NVERT) return;
    int c = (i / NVERT) % COUTC;
    float a = fmaf(x1[i],  st3[2 * c],  st3[2 * c + 1]);
    float b = fmaf(y1b[i], st1b[2 * c], st1b[2 * c + 1]);
    float v = a + b;
    out[i] = v > 0.f ? v : 0.f;
}

// ---------------------------------------------------------------------------
extern "C" void kernel_launch(void* const* d_in, const int* in_sizes, int n_in,
                              void* d_out, int out_size, void* d_ws, size_t ws_size,
                              hipStream_t stream)
{
    const float* x      = (const float*)d_in[0];
    const float* w1a    = (const float*)d_in[1];
    const float* b1a    = (const float*)d_in[2];
    const float* g1a    = (const float*)d_in[3];
    const float* be1a   = (const float*)d_in[4];
    const float* coeffs = (const float*)d_in[5];
    const float* g2a    = (const float*)d_in[6];
    const float* be2a   = (const float*)d_in[7];
    const float* w3a    = (const float*)d_in[8];
    const float* b3a    = (const float*)d_in[9];
    const float* g3a    = (const float*)d_in[10];
    const float* be3a   = (const float*)d_in[11];
    const float* w1b    = (const float*)d_in[12];
    const float* b1b    = (const float*)d_in[13];
    const float* g1b    = (const float*)d_in[14];
    const float* be1b   = (const float*)d_in[15];
    const float* Gv     = (const float*)d_in[16];
    const float* Lv     = (const float*)d_in[17];
    const float* Fv     = (const float*)d_in[18];
    const float* EW     = (const float*)d_in[19];
    const float* NS     = (const float*)d_in[20];
    const int*   Gc     = (const int*)d_in[22];   // rows are repeat(arange): ELL
    const int*   Lc     = (const int*)d_in[24];
    const int*   Fc     = (const int*)d_in[26];
    (void)in_sizes; (void)n_in; (void)out_size; (void)ws_size;

    float* ws = (float*)d_ws;
    const size_t P  = (size_t)BATCH * CNECK * NVERT;   // 10,487,808
    const size_t PF = (size_t)BATCH * CNECK * NFACE;   // 20,971,520
    float* feat  = ws;            // panels: [h | lap | gv_ew | gv_ns], 4*P
    float* y1b   = ws + 4 * P;    // 2*P (shortcut conv output)
    float* gfreg = ws + 6 * P;    // 2*PF; later reused: out2 (P) + x1 (2*P)
    float* gf_ew = gfreg;
    float* gf_ns = gfreg + PF;
    float* out2  = gfreg;         // safe: gf consumed before gemm2 writes
    float* x1    = gfreg + P;
    float* st    = ws + 6 * P + 2 * PF;
    float* st1a = st;             // 2*128
    float* st1b = st + 256;       // 2*256
    float* st2a = st + 768;       // 2*128
    float* st3a = st + 1024;      // 2*256

    const dim3 blk(256);
    const int NT = (NVERT + 63) / 64;                 // 161 N-tiles
    const int tot1 = BATCH * CNECK * NVERT;
    const int totf = BATCH * CNECK * NFACE;
    const int tot2 = BATCH * COUTC * NVERT;

    // 1) y1a = w1a @ x + b1a  -> feat panel 0
    gemm_wmma_f32<<<dim3(NT, CNECK / 128, BATCH), blk, 0, stream>>>(
        w1a, (long)CIN, 1L, 0L, x, 0L, b1a, feat, CNECK, NVERT, 1);
    // 2) y1b = w1b @ x + b1b  (shortcut, BN deferred to epilogue)
    gemm_wmma_f32<<<dim3(NT, COUTC / 128, BATCH), blk, 0, stream>>>(
        w1b, (long)CIN, 1L, 0L, x, 0L, b1b, y1b, COUTC, NVERT, 1);
    // 3) bn1a stats + apply relu in place -> h
    bn_stats<<<dim3(CNECK), blk, 0, stream>>>(feat, CNECK, NVERT, g1a, be1a, st1a);
    bn_apply_relu<<<dim3((tot1 + 255) / 256), blk, 0, stream>>>(feat, st1a, CNECK, NVERT, tot1);
    // 4) face gradients fused with EW/NS dot
    face_grad<<<dim3((totf + 255) / 256), blk, 0, stream>>>(feat, Gc, Gv, EW, NS, gf_ew, gf_ns);
    // 5) laplacian + F2V -> feat panels 1..3
    vertex_feats<<<dim3((tot1 + 255) / 256), blk, 0, stream>>>(
        feat, Lc, Lv, Fc, Fv, gf_ew, gf_ns, feat + P, feat + 2 * P, feat + 3 * P);
    // 6) out2 = sum_j coeffs_j @ feat_j ; coeffs_j[o,c] = coeffs[(c*4+j)*128 + o]
    gemm_wmma_f32<<<dim3(NT, 1, BATCH), blk, 0, stream>>>(
        coeffs, 1L, 512L, 128L, feat, (long)P, nullptr, out2, CNECK, NVERT, 4);
    // 7) bn2a + relu in place -> h2
    bn_stats<<<dim3(CNECK), blk, 0, stream>>>(out2, CNECK, NVERT, g2a, be2a, st2a);
    bn_apply_relu<<<dim3((tot1 + 255) / 256), blk, 0, stream>>>(out2, st2a, CNECK, NVERT, tot1);
    // 8) x1 = w3a @ h2 + b3a
    gemm_wmma_f32<<<dim3(NT, COUTC / 128, BATCH), blk, 0, stream>>>(
        w3a, (long)CNECK, 1L, 0L, out2, 0L, b3a, x1, COUTC, NVERT, 1);
    // 9) stats for both BNs feeding the residual epilogue
    bn_stats<<<dim3(COUTC), blk, 0, stream>>>(x1,  COUTC, NVERT, g3a, be3a, st3a);
    bn_stats<<<dim3(COUTC), blk, 0, stream>>>(y1b, COUTC, NVERT, g1b, be1b, st1b);
    // 10) out = relu(bn3a(x1) + bn1b(y1b))
    final_res<<<dim3((tot2 + 255) / 256), blk, 0, stream>>>(x1, y1b, st3a, st1b, (float*)d_out);
}